// EqualtimeLayer_89120571392335
// MI455X (gfx1250) — compile-verified
//
#include <hip/hip_runtime.h>
#include <math.h>

// CDNA5 WMMA vector types (wave32: 16x16 f32 C/D = 8 VGPRs, 16x4 f32 A = 2 VGPRs)
typedef __attribute__((ext_vector_type(2))) float v2f;
typedef __attribute__((ext_vector_type(8))) float v8f;

#define N_IN  512
#define N_OUT 256
#define NB    128
#define INV_E 0.36787944117144233f

// fast hardware reciprocal (v_rcp_f32, ~1 ulp) -- plenty for a Halley-refined solve
__device__ __forceinline__ float frcp(float x) { return __builtin_amdgcn_rcpf(x); }

__device__ __forceinline__ float lambertw0_dev(float z) {
  // principal branch, z >= -1/e + eps; piecewise init + Halley (cubic conv.)
  const float e = 2.7182818284590452f;
  float nearr = -1.f + __builtin_amdgcn_sqrtf(fmaxf(2.f * (e * z + 1.f), 0.f));
  float lz    = __logf(fmaxf(z, 1e-30f));
  float large = lz - __logf(fmaxf(lz, 1e-30f));
  float w = (z < 0.25f) ? nearr : ((z > e) ? large : __logf(1.f + fmaxf(z, -0.99f)));
#pragma unroll
  for (int i = 0; i < 4; ++i) {
    float ew  = __expf(w);
    float f   = fmaf(w, ew, -z);
    float wp1 = w + 1.f;
    float den = ew * wp1 - (w + 2.f) * f * frcp(2.f * wp1);
    den = (fabsf(den) < 1e-12f) ? 1e-12f : den;
    w = w - f * frcp(den);
  }
  return w;
}

__global__ __launch_bounds__(256)
void ttfs_equaltime_forward(const float* __restrict__ S,   // [B, N_IN] spike times
                            const float* __restrict__ W,   // [N_IN, N_OUT] weights
                            float* __restrict__ out) {     // [B, N_OUT]
  __shared__ float skey[N_IN];   // sorted spike times (original values, may be +inf)
  __shared__ int   sidx[N_IN];   // argsort indices
  __shared__ float ewL[N_IN];    // exp(t) masked (0 for inf spikes)
  __shared__ float tewL[N_IN];   // t*exp(t) masked

  const int b    = blockIdx.x;
  const int tid  = threadIdx.x;
  const int lane = tid & 31;
  const int half = lane >> 4;    // 0: lanes 0-15, 1: lanes 16-31
  const int m    = lane & 15;    // row (A) / column (B,C,D) index within tile
  const int wave = tid >> 5;     // 8 waves per block

  // ---- Phase 1: load + bitonic argsort (ascending) of 512 keys, 256 threads ----
  skey[tid]       = S[b * N_IN + tid];        sidx[tid]       = tid;
  skey[tid + 256] = S[b * N_IN + tid + 256];  sidx[tid + 256] = tid + 256;
  __syncthreads();
  for (int k = 2; k <= N_IN; k <<= 1) {
    for (int j = k >> 1; j > 0; j >>= 1) {
      int  i  = 2 * tid - (tid & (j - 1));   // lower index of this thread's pair
      int  p  = i + j;
      bool up = ((i & k) == 0);
      float ka = skey[i], kb2 = skey[p];
      bool sw = up ? (ka > kb2) : (ka < kb2);
      if (sw) {
        skey[i] = kb2; skey[p] = ka;
        int t0 = sidx[i]; sidx[i] = sidx[p]; sidx[p] = t0;
      }
      __syncthreads();
    }
  }
  // ---- Phase 2: masked exponent terms ----
  for (int n = tid; n < N_IN; n += 256) {
    float s   = skey[n];
    bool  bad = __builtin_isinf(s);
    float sm  = bad ? 0.f : s;
    float ex  = bad ? 0.f : __expf(sm);  // zero weight row => term = 0 (matches ref mask)
    ewL[n]  = ex;
    tewL[n] = sm * ex;
  }
  __syncthreads();

  // ---- tril(16x16) A-matrix chunks for the prefix-scan matmul (kb-invariant) ----
  // A layout (16x4 f32): lanes 0-15 -> K = v (VGPR v), lanes 16-31 -> K = v+2.
  v2f A[4];
#pragma unroll
  for (int c = 0; c < 4; ++c)
#pragma unroll
    for (int v = 0; v < 2; ++v)
      A[c][v] = ((4 * c + v + 2 * half) <= m) ? 1.f : 0.f;

  // ---- Phase 3: per o-tile scan + Lambert solve ----
#pragma unroll
  for (int tt = 0; tt < 2; ++tt) {
    const int ot = wave + 8 * tt;        // o-tile 0..15
    const int oG = ot * 16 + m;          // this lane's output column
    float carryA = 0.f, carryB = 0.f;    // prefix carries for column oG
    float mn = __builtin_inff();

    for (int kb = 0; kb < N_IN / 16; ++kb) {
      v8f dA = {0.f,0.f,0.f,0.f,0.f,0.f,0.f,0.f};
      v8f dB = {0.f,0.f,0.f,0.f,0.f,0.f,0.f,0.f};
#pragma unroll
      for (int c = 0; c < 4; ++c) {
        // B layout (4x16 f32): VGPR v, lane -> K = v + 2*half, N = m
        int   j0 = kb * 16 + 4 * c + 2 * half;
        int   r0 = sidx[j0], r1 = sidx[j0 + 1];
        float w0 = W[r0 * N_OUT + oG];
        float w1 = W[r1 * N_OUT + oG];
        v2f b1 = { ewL[j0]  * w0, ewL[j0 + 1]  * w1 };   // e^t * w
        v2f b2 = { tewL[j0] * w0, tewL[j0 + 1] * w1 };   // t e^t * w
        // inclusive within-block prefix sums via tril @ term
        dA = __builtin_amdgcn_wmma_f32_16x16x4_f32(false, A[c], false, b1,
                                                   (short)0, dA, false, false);
        dB = __builtin_amdgcn_wmma_f32_16x16x4_f32(false, A[c], false, b2,
                                                   (short)0, dB, false, false);
      }
      // C/D layout: VGPR r, lanes 0-15 -> k_local=r, lanes 16-31 -> k_local=r+8
#pragma unroll
      for (int r = 0; r < 8; ++r) {
        int   k   = kb * 16 + r + 8 * half;
        float ssk = skey[k];
        float nxt = (k == N_IN - 1) ? __builtin_inff() : skey[k + 1];
        float a1  = dA[r] + carryA;
        float bb  = dB[r] + carryB;
        float a1s = (fabsf(a1) < 1e-10f) ? 1e-10f : a1;
        float inv_a1 = frcp(a1s);                  // one v_rcp shared by ratio and z
        float ratio  = bb * inv_a1;
        float z      = -__expf(ratio) * inv_a1;    // C_FACTOR = 1, tau = 1
        // exact pre-window prune (no W needed):
        //   W0(z) >= -1           => t = ratio - W <= ratio + 1; t < ssk is certain if ratio+1 < ssk
        //   z < 0 => W0(z) < 0    => t > ratio;     t > nxt is certain if ratio > nxt
        bool valid = (z >= -INV_E);
        bool need  = valid && (ratio + 1.f >= ssk) && !((z < 0.f) && (ratio > nxt));
        float t = __builtin_inff();
        if (__ballot(need)) {                      // wave-uniform skip of Lambert W
          float zs   = fmaxf(z, -INV_E + 1e-7f);
          float wlam = lambertw0_dev(zs);
          t = need ? (ratio - wlam) : __builtin_inff();
        }
        // causality window: t_k <= t <= t_{k+1} (last k has no upper bound)
        if (t < ssk || t > nxt) t = __builtin_inff();
        mn = fminf(mn, t);
      }
      // carry = full-block column sum = row M=15 -> VGPR7 of lane 16+o
      carryA += __shfl(dA[7], 16 + m, 32);
      carryB += __shfl(dB[7], 16 + m, 32);
    }
    // combine the two lane-halves (k_local 0-7 vs 8-15) and store
    mn = fminf(mn, __shfl_xor(mn, 16, 32));
    if (half == 0) out[b * N_OUT + oG] = mn;
  }
}

extern "C" void kernel_launch(void* const* d_in, const int* in_sizes, int n_in,
                              void* d_out, int out_size, void* d_ws, size_t ws_size,
                              hipStream_t stream) {
  const float* spikes  = (const float*)d_in[0];  // [128, 512] f32
  const float* weights = (const float*)d_in[1];  // [512, 256] f32
  float*       out     = (float*)d_out;          // [128, 256] f32
  (void)in_sizes; (void)n_in; (void)out_size; (void)d_ws; (void)ws_size;
  ttfs_equaltime_forward<<<NB, 256, 0, stream>>>(spikes, weights, out);
}